// MCPBRNN_Generic_PETconstraint_Scaling_BYPASSM0_10196252360839
// MI455X (gfx1250) — compile-verified
//
#include <hip/hip_runtime.h>

// ---------------------------------------------------------------------------
// MCPBRNN scalar-recurrence (H=1, S=1) for MI455X / gfx1250.
//   K0: derive scalar params (single thread)
//   K1: parallel precompute of ol[b]            (memory-bound)
//   K2: WMMA-f32 based partial sum/sumsq of y_obs[100:200000]
//   K3: final std (ddof=1)
//   K4: sequential carry chain (1 thread, exp2/rcp fast path + prefetch)
//   K5: parallel finalize of all 13*B outputs
// ---------------------------------------------------------------------------

#define BTOT     262144
#define SPIN     100
#define TRAINL   200000
#define NRED     (TRAINL - SPIN)        // 199900
#define ML_C     2.9086f
#define SL_C     1.898f

// workspace float layout
#define WS_SUM   0     // 128 block partial sums
#define WS_SQ    128   // 128 block partial sums of squares
#define WS_STD   256   // final obsstd scalar
#define WS_P     260   // derived params: thr,oo1,ol1,k1o,k0o,k1l,k0l

typedef __attribute__((ext_vector_type(2))) float v2f;
typedef __attribute__((ext_vector_type(8))) float v8f;

__device__ __forceinline__ float fexp(float z) {
  return __builtin_amdgcn_exp2f(z * 1.44269504088896340736f);
}
__device__ __forceinline__ float fsig(float z) {   // 1/(1+exp(-z))
  float e = __builtin_amdgcn_exp2f(z * -1.44269504088896340736f);
  return __builtin_amdgcn_rcpf(1.0f + e);
}

// ---------------- K0: derive scalar parameters ------------------------------
__global__ void k0_params(const float* cmean, const float* cstd,
                          const float* W_rom, const float* W_rlm, const float* W_rfm,
                          const float* b0_yom, const float* W_b1_yom,
                          const float* b0_ylm, const float* W_b2_ylm,
                          const float* theltaC, float* wsf) {
  float e_o = fexp(W_rom[0]);
  float e_l = fexp(W_rlm[0]);
  float e_f = fexp(W_rfm[0]);
  float denom = e_o + e_l + e_f;
  float oo1 = e_o / denom;
  float ol1 = e_l / denom;
  float thr = fexp(theltaC[0]);
  float k1o = W_b1_yom[0] / cstd[0];
  float k0o = b0_yom[0] - cmean[0] * k1o;
  float k1l = W_b2_ylm[0] / SL_C;
  float k0l = b0_ylm[0] - ML_C * k1l;
  wsf[WS_P + 0] = thr;
  wsf[WS_P + 1] = oo1;
  wsf[WS_P + 2] = ol1;
  wsf[WS_P + 3] = k1o;
  wsf[WS_P + 4] = k0o;
  wsf[WS_P + 5] = k1l;
  wsf[WS_P + 6] = k0l;
}

// ---------------- K1: ol[b] = ol1 * sigmoid(k0l + k1l*u2[b]) ----------------
// staged into the g_ol output slot (offset 7*B); K5 re-reads and masks it.
__global__ void k1_preol(const float* __restrict__ x,
                         const float* __restrict__ wsf,
                         float* __restrict__ out) {
  int b = blockIdx.x * blockDim.x + threadIdx.x;
  if (b >= BTOT) return;
  float ol1 = wsf[WS_P + 2];
  float k1l = wsf[WS_P + 5];
  float k0l = wsf[WS_P + 6];
  float u2 = x[2 * b + 1];
  out[7 * BTOT + b] = ol1 * fsig(k0l + k1l * u2);
}

// ---------------- K2: WMMA-based partial reduction of y_obs -----------------
// D = A(16x4 f32) * ones(4x16) + C : every D column holds the A row sums.
// Each wave folds 64 f32 values per WMMA pair (sum and sum-of-squares).
__global__ void k2_reduce(const float* __restrict__ y, float* __restrict__ wsf) {
  __shared__ float red[512];
  const int tid = threadIdx.x;
  const int gid = blockIdx.x * blockDim.x + tid;      // 128*256 = 32768 threads

  v8f acc  = {};
  v8f acc2 = {};
  v2f ones; ones[0] = 1.0f; ones[1] = 1.0f;

#pragma unroll
  for (int r = 0; r < 4; ++r) {                        // 4*65536 >= 199900
    int idx = SPIN + (r * 32768 + gid) * 2;
    v2f a;
    a[0] = (idx     < TRAINL) ? y[idx]     : 0.0f;
    a[1] = (idx + 1 < TRAINL) ? y[idx + 1] : 0.0f;
    v2f aa; aa[0] = a[0] * a[0]; aa[1] = a[1] * a[1];
    // v_wmma_f32_16x16x4_f32 : (neg_a, A, neg_b, B, c_mod, C, reuse_a, reuse_b)
    acc  = __builtin_amdgcn_wmma_f32_16x16x4_f32(false, a,  false, ones,
                                                 (short)0, acc,  false, false);
    acc2 = __builtin_amdgcn_wmma_f32_16x16x4_f32(false, aa, false, ones,
                                                 (short)0, acc2, false, false);
  }

  // per-lane fold of the 8 accumulator rows (each column duplicated 16x)
  float ps = 0.f, pq = 0.f;
#pragma unroll
  for (int i = 0; i < 8; ++i) { ps += acc[i]; pq += acc2[i]; }

  red[tid] = ps;
  red[256 + tid] = pq;
  __syncthreads();
  for (int s = 128; s > 0; s >>= 1) {
    if (tid < s) {
      red[tid] += red[tid + s];
      red[256 + tid] += red[256 + tid + s];
    }
    __syncthreads();
  }
  if (tid == 0) {
    // every element was counted 16x (16 identical D columns)
    wsf[WS_SUM + blockIdx.x] = red[0]   * (1.0f / 16.0f);
    wsf[WS_SQ  + blockIdx.x] = red[256] * (1.0f / 16.0f);
  }
}

// ---------------- K3: final std with ddof=1 ---------------------------------
__global__ void k3_std(float* wsf) {
  __shared__ float red[256];
  int tid = threadIdx.x;                               // 128 threads
  red[tid]       = wsf[WS_SUM + tid];
  red[128 + tid] = wsf[WS_SQ + tid];
  __syncthreads();
  for (int s = 64; s > 0; s >>= 1) {
    if (tid < s) { red[tid] += red[tid + s]; red[128 + tid] += red[128 + tid + s]; }
    __syncthreads();
  }
  if (tid == 0) {
    float S = red[0], Q = red[128];
    const float N = (float)NRED;
    float var = (Q - S * S / N) / (N - 1.0f);
    wsf[WS_STD] = sqrtf(fmaxf(var, 0.0f));
  }
}

// ---------------- K4: the irreducible sequential carry chain ----------------
__global__ void k4_scan(const float* __restrict__ x,
                        const float* __restrict__ wsf,
                        const int* __restrict__ tlp,
                        float* __restrict__ out) {
  const float thr = wsf[WS_P + 0];
  const float oo1 = wsf[WS_P + 1];
  const float k1o = wsf[WS_P + 3];
  const float k0o = wsf[WS_P + 4];
  const int   tl  = tlp[0];
  const float* __restrict__ olp = out + 7 * BTOT;   // pre-ol from K1
  float* __restrict__ cout = out + BTOT;            // entry-carry into c_n slot

  float c = 0.0f;
  for (int b = 0; b < BTOT; ++b) {
    __builtin_prefetch(x + 2 * b + 512, 0, 0);       // global_prefetch_b8
    __builtin_prefetch(olp + b + 256, 0, 0);
    float u1 = x[2 * b];
    float u2 = x[2 * b + 1];
    float ol = olp[b];

    cout[b] = c;                                     // carry at step entry

    float px = fmaxf(u1 + c - thr, 0.0f);
    float oo = oo1 * fsig(k0o + k1o * c);
    float olc = ol;
    if (c > 0.0f) {
      float r = fmaxf(ol - u2 * __builtin_amdgcn_rcpf(c), 0.0f);
      olc = ol - r;
    }
    float f = 1.0f - oo - olc;
    float c1 = f * c + (u1 - px);
    c = (b >= tl) ? c1 : c;
  }
}

// ---------------- K5: parallel finalize of all outputs ----------------------
__global__ void k5_final(const float* __restrict__ x,
                         const float* __restrict__ wsf,
                         const int* __restrict__ tlp,
                         float* __restrict__ out) {
  int b = blockIdx.x * blockDim.x + threadIdx.x;
  if (b >= BTOT) return;

  const float thr = wsf[WS_P + 0];
  const float oo1 = wsf[WS_P + 1];
  const float k1o = wsf[WS_P + 3];
  const float k0o = wsf[WS_P + 4];
  const float obsstd = wsf[WS_STD];
  const int   tl  = tlp[0];

  float u1 = x[2 * b];
  float u2 = x[2 * b + 1];
  float c  = out[1 * BTOT + b];   // entry carry from K4
  float ol = out[7 * BTOT + b];   // pre-ol from K1

  float px = fmaxf(u1 + c - thr, 0.0f);
  float ib = (u1 > 0.0f) ? px * __builtin_amdgcn_rcpf(u1) : 0.0f;
  float oo = oo1 * fsig(k0o + k1o * c);
  float olc = ol;
  if (c > 0.0f) {
    float r = fmaxf(ol - u2 * __builtin_amdgcn_rcpf(c), 0.0f);
    olc = ol - r;
  }
  float f = 1.0f - oo - olc;
  float h = oo * c + px;
  float m = (b >= tl) ? 1.0f : 0.0f;

  out[0 * BTOT + b]  = m * h;
  out[1 * BTOT + b]  = m * c;
  out[2 * BTOT + b]  = m * (ol * c);
  out[3 * BTOT + b]  = m * (olc * c);
  out[4 * BTOT + b]  = m * px;
  out[5 * BTOT + b]  = m * ib;
  out[6 * BTOT + b]  = m * oo;
  out[7 * BTOT + b]  = m * ol;
  out[8 * BTOT + b]  = m * olc;
  out[9 * BTOT + b]  = m * f;
  out[10 * BTOT + 2 * b]     = m * h;        // h_nout[:,0]
  out[10 * BTOT + 2 * b + 1] = m * obsstd;   // h_nout[:,1]
  out[12 * BTOT + b] = m * obsstd;           // obs_std
}

// ---------------------------------------------------------------------------
extern "C" void kernel_launch(void* const* d_in, const int* in_sizes, int n_in,
                              void* d_out, int out_size, void* d_ws, size_t ws_size,
                              hipStream_t stream) {
  const float* x        = (const float*)d_in[0];
  const float* y_obs    = (const float*)d_in[1];
  const float* cmean    = (const float*)d_in[2];
  const float* cstd     = (const float*)d_in[3];
  const float* W_rom    = (const float*)d_in[4];
  const float* W_rlm    = (const float*)d_in[5];
  const float* W_rfm    = (const float*)d_in[6];
  const float* b0_yom   = (const float*)d_in[7];
  const float* W_b1_yom = (const float*)d_in[8];
  const float* b0_ylm   = (const float*)d_in[9];
  const float* W_b2_ylm = (const float*)d_in[10];
  const float* theltaC  = (const float*)d_in[11];
  const int*   time_lag = (const int*)d_in[13];
  float* out = (float*)d_out;
  float* wsf = (float*)d_ws;

  k0_params<<<1, 1, 0, stream>>>(cmean, cstd, W_rom, W_rlm, W_rfm,
                                 b0_yom, W_b1_yom, b0_ylm, W_b2_ylm,
                                 theltaC, wsf);
  k1_preol<<<BTOT / 256, 256, 0, stream>>>(x, wsf, out);
  k2_reduce<<<128, 256, 0, stream>>>(y_obs, wsf);
  k3_std<<<1, 128, 0, stream>>>(wsf);
  k4_scan<<<1, 1, 0, stream>>>(x, wsf, time_lag, out);
  k5_final<<<BTOT / 256, 256, 0, stream>>>(x, wsf, time_lag, out);
}